// GraphSAGEThreeLayer_25460566130890
// MI455X (gfx1250) — compile-verified
//
#include <hip/hip_runtime.h>
#include <hip/hip_bf16.h>

typedef float v2f __attribute__((ext_vector_type(2)));
typedef float v8f __attribute__((ext_vector_type(8)));
typedef int   v4i __attribute__((ext_vector_type(4)));

#define IN_C 256
#define H1 256
#define H2 128
#define H3 64
#define OUT_C 2

#define GAS __attribute__((address_space(1)))
#define LAS __attribute__((address_space(3)))

// ---------------- CDNA5 async global->LDS staging (ASYNCcnt path) ----------------
__device__ __forceinline__ void async_copy_b128(const float* gsrc, float* ldst) {
#if __has_builtin(__builtin_amdgcn_global_load_async_to_lds_b128)
    __builtin_amdgcn_global_load_async_to_lds_b128(
        (GAS v4i*)(uintptr_t)gsrc, (LAS v4i*)ldst, 0, 0);
#else
    *(float4*)ldst = *(const float4*)gsrc;   // sync fallback
#endif
}
__device__ __forceinline__ void async_copy_b32(const float* gsrc, float* ldst) {
#if __has_builtin(__builtin_amdgcn_global_load_async_to_lds_b32)
    __builtin_amdgcn_global_load_async_to_lds_b32(
        (GAS int*)(uintptr_t)gsrc, (LAS int*)ldst, 0, 0);
#else
    *ldst = *gsrc;                           // sync fallback
#endif
}
__device__ __forceinline__ void wait_async0() {
#if __has_builtin(__builtin_amdgcn_s_wait_asynccnt)
    __builtin_amdgcn_s_wait_asynccnt(0);
#endif
}

// ---------------- utility: zero a float buffer (grid-stride, float4) ----------------
__global__ void zero_f4_kernel(float4* __restrict__ p, long long n4) {
    long long i = blockIdx.x * (long long)blockDim.x + threadIdx.x;
    long long stride = (long long)gridDim.x * blockDim.x;
    float4 z; z.x = 0.f; z.y = 0.f; z.z = 0.f; z.w = 0.f;
    for (; i < n4; i += stride) p[i] = z;
}

// ---------------- edge prep: int64 -> int32, integer degree atomics ----------------
__global__ void edge_prep_kernel(const long long* __restrict__ ei,
                                 int* __restrict__ src32, int* __restrict__ dst32,
                                 int* __restrict__ deg, int E) {
    int e = blockIdx.x * blockDim.x + threadIdx.x;
    if (e >= E) return;
    int s = (int)ei[e];                 // row 0: src
    int d = (int)ei[(long long)E + e];  // row 1: dst
    src32[e] = s;
    dst32[e] = d;
    atomicAdd(&deg[d], 1);
}

// inv_deg[i] = 1 / max(deg[i], 1)   (written in place over the int degree buffer)
__global__ void finalize_deg_kernel(int* __restrict__ deg, int N) {
    int i = blockIdx.x * blockDim.x + threadIdx.x;
    if (i >= N) return;
    int d = deg[i];
    float inv = 1.0f / (float)(d > 1 ? d : 1);
    ((float*)deg)[i] = inv;
}

// ---------------- aggregation: agg[dst] += x[src] * inv_deg[dst] ----------------
// chunk index fastest-varying -> coalesced float4 gathers; native f32 atomics hit L2
// (per-layer agg buffer <= 102 MB, fully resident in the 192 MB L2)
__global__ void aggregate_kernel(const float* __restrict__ x,
                                 const int* __restrict__ src, const int* __restrict__ dst,
                                 const float* __restrict__ inv_deg,
                                 float* __restrict__ agg, int E, int F) {
    long long t = blockIdx.x * (long long)blockDim.x + threadIdx.x;
    int chunks = F >> 2;
    long long total = (long long)E * chunks;
    if (t >= total) return;
    int e = (int)(t / chunks);
    int c = (int)(t % chunks);
    int s = src[e];
    int d = dst[e];
    float w = inv_deg[d];
    const float4 v = *(const float4*)(x + (long long)s * F + c * 4);
    float* base = agg + (long long)d * F + c * 4;
    unsafeAtomicAdd(base + 0, v.x * w);
    unsafeAtomicAdd(base + 1, v.y * w);
    unsafeAtomicAdd(base + 2, v.z * w);
    unsafeAtomicAdd(base + 3, v.w * w);
}

// ---------------- fused SAGE layer GEMM via V_WMMA_F32_16X16X4_F32 ----------------
// out[N,F] = relu( A0[N,K] @ W0[K,F] + A1[N,K] @ W1[K,F] + bias )
// 128x64 block tile, BK=32, double-buffered LDS filled by async global->LDS;
// 8 waves, each owning a 32x32 region (2x2 WMMA tiles). B stored transposed in LDS
// so all four fragments are single aligned ds_load_b64s.
#define BM 128
#define BN 64
#define BK 32

__global__ __launch_bounds__(256) void sage_gemm_kernel(
        const float* __restrict__ A0, const float* __restrict__ A1,
        const float* __restrict__ W0, const float* __restrict__ W1,
        const float* __restrict__ bias, float* __restrict__ out,
        int N, int K, int F, int relu) {
    // stride 36 floats = 144 B: multiple of 16 (b128 async stores, float4) and
    // even (b64 fragment loads at even k offsets)
    __shared__ __attribute__((aligned(16))) float As[2][BM][36];
    __shared__ __attribute__((aligned(16))) float Bs[2][BN][36];  // transposed [col][k]

    const int t     = threadIdx.x;
    const int lane  = t & 31;
    const int wave  = t >> 5;       // 0..7
    const int waveM = wave & 3;     // 0..3 -> 32-row group
    const int waveN = wave >> 2;    // 0..1 -> 32-col group
    const int m0    = lane & 15;
    const int khalf = lane >> 4;    // 0/1

    const int rowBase = blockIdx.x * BM;
    const int colBase = blockIdx.y * BN;
    const int nchunks = (2 * K) / BK;

    // stage one BK-chunk into LDS buffer `buf` (A: b128/lane, B: b32 transposed)
    auto stage = [&](int buf, int chunk) {
        const int kk = chunk * BK;
        const float* Asrc = (kk < K) ? A0 : A1;
        const float* Wsrc = (kk < K) ? W0 : W1;
        const int kloc = (kk < K) ? kk : (kk - K);
        // A tile 128x32: 1024 b128 transfers, 4 per thread.
        // OOB rows clamped: garbage rows only affect never-stored output rows.
#pragma unroll
        for (int i = 0; i < 4; ++i) {
            int fid = t + i * 256;   // 0..1023
            int r   = fid >> 3;      // 0..127
            int c4  = fid & 7;       // 0..7
            int rowg = rowBase + r;
            if (rowg > N - 1) rowg = N - 1;
            async_copy_b128(Asrc + (long long)rowg * K + kloc + c4 * 4,
                            &As[buf][r][c4 * 4]);
        }
        // B tile 32x64 -> LDS transposed [col][k]: 2048 b32 transfers, 8 per thread.
        // Consecutive lanes read consecutive cols -> coalesced global reads.
#pragma unroll
        for (int i = 0; i < 8; ++i) {
            int fid = t + i * 256;   // 0..2047
            int col = fid & 63;
            int k   = fid >> 6;      // 0..31
            async_copy_b32(Wsrc + (long long)(kloc + k) * F + colBase + col,
                           &Bs[buf][col][k]);
        }
    };

    v8f acc[2][2] = {};

    stage(0, 0);
    for (int c = 0; c < nchunks; ++c) {
        wait_async0();        // my transfers for chunk c have landed in LDS
        __syncthreads();      // everyone's landed; prior reads of other buf done
        if (c + 1 < nchunks) stage((c + 1) & 1, c + 1);  // overlap with compute
        const int buf = c & 1;
#pragma unroll
        for (int kb = 0; kb < 8; ++kb) {
            const int kf = kb * 4 + 2 * khalf;
            // A fragments (ISA: lane m=L%16 holds ks 2*(L/16)+{0,1})
            v2f a0 = *(const v2f*)&As[buf][waveM * 32 +  0 + m0][kf];
            v2f a1 = *(const v2f*)&As[buf][waveM * 32 + 16 + m0][kf];
            // B fragments from transposed tile: single b64 per fragment
            v2f b0 = *(const v2f*)&Bs[buf][waveN * 32 +  0 + m0][kf];
            v2f b1 = *(const v2f*)&Bs[buf][waveN * 32 + 16 + m0][kf];

            acc[0][0] = __builtin_amdgcn_wmma_f32_16x16x4_f32(false, a0, false, b0,
                            (short)0, acc[0][0], false, false);
            acc[0][1] = __builtin_amdgcn_wmma_f32_16x16x4_f32(false, a0, false, b1,
                            (short)0, acc[0][1], false, false);
            acc[1][0] = __builtin_amdgcn_wmma_f32_16x16x4_f32(false, a1, false, b0,
                            (short)0, acc[1][0], false, false);
            acc[1][1] = __builtin_amdgcn_wmma_f32_16x16x4_f32(false, a1, false, b1,
                            (short)0, acc[1][1], false, false);
        }
        __syncthreads();      // all reads of `buf` done before it is refilled
    }

    // epilogue: bias + relu, C/D layout row = v + 8*(L/16), col = L%16
#pragma unroll
    for (int tm = 0; tm < 2; ++tm) {
#pragma unroll
        for (int tn = 0; tn < 2; ++tn) {
#pragma unroll
            for (int v = 0; v < 8; ++v) {
                int row = rowBase + waveM * 32 + tm * 16 + v + 8 * khalf;
                int col = colBase + waveN * 32 + tn * 16 + m0;
                if (row < N) {
                    float val = acc[tm][tn][v] + bias[col];
                    if (relu) val = fmaxf(val, 0.f);
                    out[(long long)row * F + col] = val;
                }
            }
        }
    }
}

// ---------------- tiny classifier: out[N,2] = h[N,64] @ Wc + bc ----------------
__global__ void classifier_kernel(const float* __restrict__ h, const float* __restrict__ Wc,
                                  const float* __restrict__ bc, float* __restrict__ out, int N) {
    __shared__ float w[H3 * OUT_C];
    if (threadIdx.x < H3 * OUT_C) w[threadIdx.x] = Wc[threadIdx.x];
    __syncthreads();
    int i = blockIdx.x * blockDim.x + threadIdx.x;
    if (i >= N) return;
    float a0 = bc[0], a1 = bc[1];
    const float* r = h + (long long)i * H3;
#pragma unroll
    for (int k = 0; k < H3; k += 4) {
        float4 v = *(const float4*)(r + k);
        a0 += v.x * w[(k + 0) * 2 + 0] + v.y * w[(k + 1) * 2 + 0]
            + v.z * w[(k + 2) * 2 + 0] + v.w * w[(k + 3) * 2 + 0];
        a1 += v.x * w[(k + 0) * 2 + 1] + v.y * w[(k + 1) * 2 + 1]
            + v.z * w[(k + 2) * 2 + 1] + v.w * w[(k + 3) * 2 + 1];
    }
    out[(long long)i * 2 + 0] = a0;
    out[(long long)i * 2 + 1] = a1;
}

// ------------------------------------------------------------------------------
extern "C" void kernel_launch(void* const* d_in, const int* in_sizes, int n_in,
                              void* d_out, int out_size, void* d_ws, size_t ws_size,
                              hipStream_t stream) {
    const float*     x  = (const float*)d_in[0];
    const long long* ei = (const long long*)d_in[1];   // int64 in reference
    const float* Wl1 = (const float*)d_in[2];
    const float* Wr1 = (const float*)d_in[3];
    const float* b1  = (const float*)d_in[4];
    const float* Wl2 = (const float*)d_in[5];
    const float* Wr2 = (const float*)d_in[6];
    const float* b2  = (const float*)d_in[7];
    const float* Wl3 = (const float*)d_in[8];
    const float* Wr3 = (const float*)d_in[9];
    const float* b3  = (const float*)d_in[10];
    const float* Wc  = (const float*)d_in[11];
    const float* bc  = (const float*)d_in[12];

    const int N = in_sizes[0] / IN_C;   // 100000
    const int E = in_sizes[1] / 2;      // 1600000

    // workspace carve-out (~270 MB): deg/inv | src32 | dst32 | agg(N*256) | hA(N*256) | hB(N*128)
    char* ws = (char*)d_ws;
    auto carve = [&](size_t bytes) {
        char* p = ws;
        ws += (bytes + 255) & ~(size_t)255;
        return p;
    };
    int*   deg   = (int*)  carve((size_t)N * 4);  // becomes inv_deg (float) in place
    int*   src32 = (int*)  carve((size_t)E * 4);
    int*   dst32 = (int*)  carve((size_t)E * 4);
    float* agg   = (float*)carve((size_t)N * 256 * 4);
    float* hA    = (float*)carve((size_t)N * 256 * 4);
    float* hB    = (float*)carve((size_t)N * 128 * 4);
    float* inv_deg = (float*)deg;

    const int ZB = 2048;  // grid-stride zero blocks

    // degrees
    zero_f4_kernel<<<64, 256, 0, stream>>>((float4*)deg, (long long)(N + 3) / 4);
    edge_prep_kernel<<<(E + 255) / 256, 256, 0, stream>>>(ei, src32, dst32, deg, E);
    finalize_deg_kernel<<<(N + 255) / 256, 256, 0, stream>>>(deg, N);

    auto run_layer = [&](const float* in, int K, const float* Wl, const float* Wr,
                         const float* b, float* outp, int F) {
        zero_f4_kernel<<<ZB, 256, 0, stream>>>((float4*)agg, (long long)N * K / 4);
        long long tot = (long long)E * (K / 4);
        aggregate_kernel<<<(int)((tot + 255) / 256), 256, 0, stream>>>(
            in, src32, dst32, inv_deg, agg, E, K);
        dim3 grid((N + BM - 1) / BM, F / BN);
        sage_gemm_kernel<<<grid, 256, 0, stream>>>(agg, in, Wl, Wr, b, outp, N, K, F, 1);
    };

    run_layer(x,  IN_C, Wl1, Wr1, b1, hA, H1);   // 256 -> 256
    run_layer(hA, H1,   Wl2, Wr2, b2, hB, H2);   // 256 -> 128
    run_layer(hB, H2,   Wl3, Wr3, b3, hA, H3);   // 128 -> 64

    classifier_kernel<<<(N + 127) / 128, 128, 0, stream>>>(hA, Wc, bc, (float*)d_out, N);
}